// SoftAttention_49898930045635
// MI455X (gfx1250) — compile-verified
//
#include <hip/hip_runtime.h>
#include <hip/hip_bf16.h>

// Problem constants (from reference)
#define BB   16
#define SS   64
#define NN   196
#define HH   1024
#define FF   2048
#define EE   512

typedef __attribute__((ext_vector_type(2))) float v2f;
typedef __attribute__((ext_vector_type(8))) float v8f;

// ---------------------------------------------------------------------------
// fp32 WMMA: V_WMMA_F32_16X16X4_F32 (CDNA5 fp32 matrix path, exact fp32).
// A (16x4): lane L holds row M=L&15; v0=A[M][k0], v1=A[M][k0+1], k0=(L<16?0:2)
// B (4x16): lane L holds col N=L&15; v0=B[k0][N], v1=B[k0+1][N]
// C/D (16x16): lane L col N=L&15, vgpr v holds row M = v + 8*(L>>4)
// ---------------------------------------------------------------------------
__device__ __forceinline__ v8f wmma_f32_step(v2f a, v2f b, v8f c) {
    return __builtin_amdgcn_wmma_f32_16x16x4_f32(
        /*neg_a=*/false, a, /*neg_b=*/false, b,
        /*c_mod=*/(short)0, c, /*reuse_a=*/false, /*reuse_b=*/false);
}

// ---------------------------------------------------------------------------
// Register-blocked fp32 GEMM: C[M,N] = A[M,K] * B[K,N].
// One wave computes an MT x NT grid of 16x16 tiles (2x2 => 32x32 output),
// amortizing operand loads: per k-step, MT b64 A-loads + 2*NT b32 B-loads
// feed MT*NT WMMAs (1.5 loads/WMMA at 2x2 vs 3.0 unblocked).
// Requires: M % (16*MT) == 0, N % (16*NT) == 0, K % 4 == 0.
// ---------------------------------------------------------------------------
template<int MT, int NT>
__global__ void gemm_f32_wmma(const float* __restrict__ A,
                              const float* __restrict__ B,
                              float* __restrict__ C,
                              int M, int N, int K) {
    const int lane = threadIdx.x & 31;
    const int wave = threadIdx.x >> 5;
    const int job  = blockIdx.x * (blockDim.x >> 5) + wave;

    const int nBlocks = N / (16 * NT);
    const int mB = job / nBlocks;
    const int nB = job % nBlocks;

    const int rowBase = mB * 16 * MT + (lane & 15);
    const int colBase = nB * 16 * NT + (lane & 15);
    const int kOff    = (lane >> 4) << 1;            // 0 or 2

    v8f acc[MT][NT];
#pragma unroll
    for (int i = 0; i < MT; ++i)
#pragma unroll
        for (int j = 0; j < NT; ++j)
            acc[i][j] = (v8f){};

    const float* ap[MT];
#pragma unroll
    for (int i = 0; i < MT; ++i)
        ap[i] = A + (size_t)(rowBase + i * 16) * K + kOff;
    const float* bp = B + (size_t)kOff * N + colBase;

    for (int k = 0; k < K; k += 4) {
        v2f afrag[MT];
#pragma unroll
        for (int i = 0; i < MT; ++i)
            afrag[i] = *(const v2f*)(ap[i] + k);     // 8B aligned (even idx)

        v2f bfrag[NT];
#pragma unroll
        for (int j = 0; j < NT; ++j) {
            bfrag[j][0] = bp[(size_t)k * N + j * 16];
            bfrag[j][1] = bp[(size_t)(k + 1) * N + j * 16];
        }

        // prefetch B rows 8 k-steps ahead (2KB row stride defeats HW streamer)
        if (k + 32 < K)
            __builtin_prefetch(bp + (size_t)(k + 32) * N, 0, 0);

#pragma unroll
        for (int i = 0; i < MT; ++i)
#pragma unroll
            for (int j = 0; j < NT; ++j)
                acc[i][j] = wmma_f32_step(afrag[i], bfrag[j], acc[i][j]);
    }

    const int rowHi = (lane >> 4) << 3;              // 0 or 8
#pragma unroll
    for (int i = 0; i < MT; ++i) {
        float* cp = C + (size_t)(mB * 16 * MT + i * 16 + rowHi) * N + colBase;
#pragma unroll
        for (int v = 0; v < 8; ++v)
#pragma unroll
            for (int j = 0; j < NT; ++j)
                cp[(size_t)v * N + j * 16] = acc[i][j][v];
    }
}

// ---------------------------------------------------------------------------
// Additive-attention energies + masked softmax.
// One block (256 threads = 8 wave32) per (b,s).  TRANS-bound: 512 tanh per
// (b,s,n); lanes stride E, reduce with 32-wide shuffles.
// ---------------------------------------------------------------------------
__global__ void scores_softmax(const float* __restrict__ Wh,     // [B*S, E]
                               const float* __restrict__ Uv,     // [B*N, E]
                               const float* __restrict__ bias,   // [E]
                               const float* __restrict__ wvec,   // [E]
                               const int*   __restrict__ amask,  // [B, N]
                               float* __restrict__ weights) {    // [B, S, N]
    __shared__ float whb[EE];     // Wh row + bias (pre-added)
    __shared__ float wsh[EE];     // w vector
    __shared__ float sc[NN];      // raw scores
    __shared__ float red[256];    // block reduction scratch

    const int bs  = blockIdx.x;          // 0 .. B*S-1
    const int b   = bs / SS;
    const int tid = threadIdx.x;

    for (int e = tid; e < EE; e += 256) {
        whb[e] = Wh[(size_t)bs * EE + e] + bias[e];
        wsh[e] = wvec[e];
    }
    __syncthreads();

    const int lane = tid & 31;
    const int wave = tid >> 5;
    for (int n = wave; n < NN; n += 8) {
        const float* uv = Uv + ((size_t)b * NN + n) * EE;
        float p = 0.f;
#pragma unroll 4
        for (int e = lane; e < EE; e += 32)
            p += wsh[e] * tanhf(whb[e] + uv[e]);
#pragma unroll
        for (int off = 16; off > 0; off >>= 1)
            p += __shfl_xor(p, off, 32);
        if (lane == 0) sc[n] = p;
    }
    __syncthreads();

    // masked softmax over N (one score per thread; N < blockDim)
    const bool act = tid < NN;
    float sval = -1e30f;
    if (act)
        sval = (amask[b * NN + tid] == 0) ? -1e9f : sc[tid];

    red[tid] = sval;
    __syncthreads();
#pragma unroll
    for (int off = 128; off > 0; off >>= 1) {
        if (tid < off) red[tid] = fmaxf(red[tid], red[tid + off]);
        __syncthreads();
    }
    const float mx = red[0];
    __syncthreads();

    const float ex = act ? __expf(sval - mx) : 0.f;
    red[tid] = ex;
    __syncthreads();
#pragma unroll
    for (int off = 128; off > 0; off >>= 1) {
        if (tid < off) red[tid] += red[tid + off];
        __syncthreads();
    }
    const float inv = 1.0f / red[0];

    if (act)
        weights[(size_t)bs * NN + tid] = ex * inv;
}

// ---------------------------------------------------------------------------
// attn_feats[b] = weights[b] (S x N) @ feats[b] (N x F), fp32 WMMA, K = 196.
// Same 2x2 register blocking; per-batch jobs = (S/32)*(F/32) = 128.
// ---------------------------------------------------------------------------
__global__ void attn_gemm(const float* __restrict__ W,      // [B, S, N]
                          const float* __restrict__ feats,  // [B, N, F]
                          float* __restrict__ out) {        // [B, S, F]
    const int lane = threadIdx.x & 31;
    const int wave = threadIdx.x >> 5;
    const int job  = blockIdx.x * (blockDim.x >> 5) + wave;

    const int jobsPerBatch = (SS / 32) * (FF / 32);  // 2 * 64 = 128
    const int b  = job / jobsPerBatch;
    const int t  = job % jobsPerBatch;
    const int mB = t / (FF / 32);
    const int nB = t % (FF / 32);

    const float* A  = W     + (size_t)b * SS * NN;   // [S, N]
    const float* Bm = feats + (size_t)b * NN * FF;   // [N, F]

    const int rowBase = mB * 32 + (lane & 15);
    const int colBase = nB * 32 + (lane & 15);
    const int kOff    = (lane >> 4) << 1;

    v8f acc[2][2];
#pragma unroll
    for (int i = 0; i < 2; ++i)
#pragma unroll
        for (int j = 0; j < 2; ++j)
            acc[i][j] = (v8f){};

    const float* ap0 = A + (size_t)rowBase * NN + kOff;        // 8B aligned
    const float* ap1 = ap0 + (size_t)16 * NN;
    const float* bp  = Bm + (size_t)kOff * FF + colBase;

#pragma unroll 7
    for (int k = 0; k < NN; k += 4) {                // 49 k-steps exactly
        v2f a0 = *(const v2f*)(ap0 + k);
        v2f a1 = *(const v2f*)(ap1 + k);
        v2f b0, b1;
        b0[0] = bp[(size_t)k * FF];
        b0[1] = bp[(size_t)(k + 1) * FF];
        b1[0] = bp[(size_t)k * FF + 16];
        b1[1] = bp[(size_t)(k + 1) * FF + 16];
        acc[0][0] = wmma_f32_step(a0, b0, acc[0][0]);
        acc[0][1] = wmma_f32_step(a0, b1, acc[0][1]);
        acc[1][0] = wmma_f32_step(a1, b0, acc[1][0]);
        acc[1][1] = wmma_f32_step(a1, b1, acc[1][1]);
    }

    const int rowHi = (lane >> 4) << 3;
#pragma unroll
    for (int i = 0; i < 2; ++i) {
        float* cp = out + ((size_t)b * SS + mB * 32 + i * 16 + rowHi) * FF + colBase;
#pragma unroll
        for (int v = 0; v < 8; ++v) {
            cp[(size_t)v * FF]      = acc[i][0][v];
            cp[(size_t)v * FF + 16] = acc[i][1][v];
        }
    }
}

extern "C" void kernel_launch(void* const* d_in, const int* in_sizes, int n_in,
                              void* d_out, int out_size, void* d_ws, size_t ws_size,
                              hipStream_t stream) {
    const float* hidden = (const float*)d_in[0];   // [B, S, H]
    const float* feats  = (const float*)d_in[1];   // [B, N, F]
    const int*   amask  = (const int*)  d_in[2];   // [B, N]
    const float* w_h    = (const float*)d_in[3];   // [H, E]
    const float* w_u    = (const float*)d_in[4];   // [F, E]
    const float* bias   = (const float*)d_in[5];   // [E]
    const float* wvec   = (const float*)d_in[6];   // [E]

    float* out     = (float*)d_out;
    float* attn    = out;                              // B*S*F
    float* weights = out + (size_t)BB * SS * FF;       // B*S*N

    float* Wh = (float*)d_ws;                          // [B*S, E]  (2 MB)
    float* Uv = Wh + (size_t)BB * SS * EE;             // [B*N, E]  (6.4 MB)

    // 1) Wh = hidden @ w_h : M=1024,N=512,K=1024 -> (32*16)=512 jobs / 8 waves
    gemm_f32_wmma<2, 2><<<(BB * SS / 32) * (EE / 32) / 8, 256, 0, stream>>>(
        hidden, w_h, Wh, BB * SS, EE, HH);

    // 2) Uv = feats @ w_u : M=3136,N=512,K=2048 -> (98*16)=1568 jobs / 8 waves
    gemm_f32_wmma<2, 2><<<(BB * NN / 32) * (EE / 32) / 8, 256, 0, stream>>>(
        feats, w_u, Uv, BB * NN, EE, FF);

    // 3) energies + masked softmax: one block per (b,s)
    scores_softmax<<<BB * SS, 256, 0, stream>>>(Wh, Uv, bias, wvec, amask, weights);

    // 4) attn = weights @ feats (batched): 2048 jobs / 8 waves
    attn_gemm<<<BB * (SS / 32) * (FF / 32) / 8, 256, 0, stream>>>(weights, feats, attn);
}